// Set2Set_58849641890192
// MI455X (gfx1250) — compile-verified
//
#include <hip/hip_runtime.h>
#include <hip/hip_bf16.h>

// Problem constants (match reference)
#define NND  500000      // nodes (also taken from in_sizes[1] at runtime)
#define DD   128         // feature dim
#define BD   2048        // graphs
#define TT   4           // processing steps
#define KD   (3 * DD)    // GEMM K   = 384  ([q | r | h])
#define NDIM (4 * DD)    // GEMM N   = 512  (i,f,g,o gates)

typedef __attribute__((ext_vector_type(2))) float v2f;
typedef __attribute__((ext_vector_type(8))) float v8f;

// ---------------------------------------------------------------------------
// Prep: zero activation buffer (q_star=0, h=0) and cell state c=0
// ---------------------------------------------------------------------------
__global__ __launch_bounds__(256) void zero_state(float* __restrict__ act,
                                                  float* __restrict__ c) {
    int idx = blockIdx.x * blockDim.x + threadIdx.x;
    if (idx < BD * KD) act[idx] = 0.0f;
    if (idx < BD * DD) c[idx] = 0.0f;
}

// ---------------------------------------------------------------------------
// Prep: pack WcT[k][n] = (k<2D ? W_ih[n][k] : W_hh[n][k-2D]); bsum = b_ih+b_hh
// ---------------------------------------------------------------------------
__global__ __launch_bounds__(256) void prep_weights(const float* __restrict__ W_ih,
                                                    const float* __restrict__ W_hh,
                                                    const float* __restrict__ b_ih,
                                                    const float* __restrict__ b_hh,
                                                    float* __restrict__ WcT,
                                                    float* __restrict__ bsum) {
    int idx = blockIdx.x * blockDim.x + threadIdx.x;
    if (idx < NDIM * KD) {
        int k = idx / NDIM, n = idx % NDIM;
        float w = (k < 2 * DD) ? W_ih[(size_t)n * (2 * DD) + k]
                               : W_hh[(size_t)n * DD + (k - 2 * DD)];
        WcT[idx] = w;
    }
    if (idx < NDIM) bsum[idx] = b_ih[idx] + b_hh[idx];
}

// ---------------------------------------------------------------------------
// Prep: segment starts from sorted batch[]. seg has BD+1 entries.
// ---------------------------------------------------------------------------
__global__ __launch_bounds__(256) void seg_bounds(const int* __restrict__ batch,
                                                  int* __restrict__ seg, int n) {
    int i = blockIdx.x * blockDim.x + threadIdx.x;
    if (i >= n) return;
    int bi = batch[i];
    if (i == 0) {
        for (int b = 0; b <= bi; ++b) seg[b] = 0;
    } else {
        int bp = batch[i - 1];
        for (int b = bp + 1; b <= bi; ++b) seg[b] = i;
    }
    if (i == n - 1) {
        for (int b = bi + 1; b <= BD; ++b) seg[b] = n;
    }
}

// ---------------------------------------------------------------------------
// LSTM input GEMM: gates[2048][512] = act[2048][384] @ WcT[384][512] + bsum
// One wave per 16x16 output tile; V_WMMA_F32_16X16X4_F32, K-loop of 96 steps.
// A fragment (ISA 7.12.2, 32-bit A 16x4): lanes 0-15 -> K=0,1; lanes 16-31 -> K=2,3.
// B fragment mirrors with N across lanes. C/D: VGPR j -> M = j + 8*(lane>=16).
// ---------------------------------------------------------------------------
__global__ __launch_bounds__(256) void gemm_gates(const float* __restrict__ act,
                                                  const float* __restrict__ WcT,
                                                  const float* __restrict__ bsum,
                                                  float* __restrict__ gates) {
    const int wave  = (blockIdx.x * blockDim.x + threadIdx.x) >> 5;   // 4096 waves
    const int lane  = threadIdx.x & 31;
    const int tn    = wave & 31;          // 32 tiles along N (512/16)
    const int tm    = wave >> 5;          // 128 tiles along M (2048/16)
    const int m0    = tm << 4;
    const int n0    = tn << 4;
    const int l16   = lane & 15;
    const int hi    = lane >> 4;          // 0: K=0,1   1: K=2,3

    const float* arow = act + (size_t)(m0 + l16) * KD;
    v8f acc = {};

#pragma unroll 4
    for (int k0 = 0; k0 < KD; k0 += 4) {
        const int ka = k0 + (hi << 1);
        v2f a = *(const v2f*)(arow + ka);                 // A[l16][ka], A[l16][ka+1]
        const float* bp = WcT + (size_t)ka * NDIM + n0 + l16;
        v2f b;
        b.x = bp[0];                                      // B[ka  ][n0+l16]
        b.y = bp[NDIM];                                   // B[ka+1][n0+l16]
        acc = __builtin_amdgcn_wmma_f32_16x16x4_f32(
            /*neg_a=*/false, a, /*neg_b=*/false, b,
            /*c_mod=*/(short)0, acc, /*reuse_a=*/false, /*reuse_b=*/false);
    }

    const int col  = n0 + l16;
    const float bi = bsum[col];
    float* gr = gates + (size_t)(m0 + (hi << 3)) * NDIM + col;
#pragma unroll
    for (int j = 0; j < 8; ++j) gr[(size_t)j * NDIM] = acc[j] + bi;
}

// ---------------------------------------------------------------------------
// Pointwise LSTM: c' = sig(f)*c + sig(i)*tanh(g);  h = sig(o)*tanh(c')
// Writes h into act[:,0:128] (q slot) and act[:,256:384] (h slot).
// ---------------------------------------------------------------------------
__device__ __forceinline__ float sigmoidf(float v) {
    return 1.0f / (1.0f + __expf(-v));
}

__global__ __launch_bounds__(256) void lstm_pointwise(const float* __restrict__ gates,
                                                      float* __restrict__ c,
                                                      float* __restrict__ act) {
    int idx = blockIdx.x * blockDim.x + threadIdx.x;
    if (idx >= BD * DD) return;
    int b = idx >> 7, d = idx & (DD - 1);
    const float* g = gates + (size_t)b * NDIM;
    float gi = g[d], gf = g[DD + d], gg = g[2 * DD + d], go = g[3 * DD + d];
    float cn = sigmoidf(gf) * c[idx] + sigmoidf(gi) * tanhf(gg);
    c[idx] = cn;
    float h = sigmoidf(go) * tanhf(cn);
    float* ar = act + (size_t)b * KD;
    ar[d]          = h;   // q part of q_star
    ar[2 * DD + d] = h;   // recurrent h input
}

// ---------------------------------------------------------------------------
// Fused per-graph attention: e_i = x_i.q ; softmax over segment ; r = sum a_i x_i
// One block (8 wave32) per graph. Wave handles a full 512B row (float4/lane).
// ---------------------------------------------------------------------------
__global__ __launch_bounds__(256) void attention_pool(const float* __restrict__ x,
                                                      const int* __restrict__ seg,
                                                      float* __restrict__ act,
                                                      float* __restrict__ ebuf) {
    __shared__ float qsh[DD];
    __shared__ float red[8];
    __shared__ float rpart[8][DD];

    const int b    = blockIdx.x;
    const int s    = seg[b];
    const int eend = seg[b + 1];
    const int cnt  = eend - s;
    const int tid  = threadIdx.x;
    const int wave = tid >> 5;
    const int lane = tid & 31;

    if (tid < DD) qsh[tid] = act[(size_t)b * KD + tid];
    __syncthreads();

    if (cnt == 0) {                       // empty graph -> r = 0
        if (tid < DD) act[(size_t)b * KD + DD + tid] = 0.0f;
        return;
    }

    const float4 qv = *(const float4*)(&qsh[lane << 2]);

    // ---- pass 1: logits + segment max -------------------------------------
    float mloc = -3.402823466e38f;
    for (int i = s + wave; i < eend; i += 8) {
        __builtin_prefetch(x + (size_t)(i + 8) * DD + (lane << 2), 0, 0);
        const float4 xv = *(const float4*)(x + (size_t)i * DD + (lane << 2));
        float p = xv.x * qv.x + xv.y * qv.y + xv.z * qv.z + xv.w * qv.w;
#pragma unroll
        for (int off = 16; off > 0; off >>= 1) p += __shfl_xor(p, off, 32);
        if (lane == 0) ebuf[i] = p;
        mloc = fmaxf(mloc, p);
    }
    if (lane == 0) red[wave] = mloc;
    __syncthreads();
    float m = red[0];
#pragma unroll
    for (int w = 1; w < 8; ++w) m = fmaxf(m, red[w]);
    __syncthreads();

    // ---- pass 2: exp-sum + weighted accumulate (x hot in L2) --------------
    float4 racc = {0.0f, 0.0f, 0.0f, 0.0f};
    float  ssum = 0.0f;
    for (int i = s + wave; i < eend; i += 8) {
        const float ex = __expf(ebuf[i] - m);
        const float4 xv = *(const float4*)(x + (size_t)i * DD + (lane << 2));
        racc.x += ex * xv.x;
        racc.y += ex * xv.y;
        racc.z += ex * xv.z;
        racc.w += ex * xv.w;
        ssum += ex;
    }
    *(float4*)(&rpart[wave][lane << 2]) = racc;
    if (lane == 0) red[wave] = ssum;
    __syncthreads();

    if (tid < DD) {
        float S = 0.0f;
#pragma unroll
        for (int w = 0; w < 8; ++w) S += red[w];
        float r = 0.0f;
#pragma unroll
        for (int w = 0; w < 8; ++w) r += rpart[w][tid];
        act[(size_t)b * KD + DD + tid] = r / S;   // r part of q_star
    }
}

// ---------------------------------------------------------------------------
// Output: q_star = act[:, 0:256]
// ---------------------------------------------------------------------------
__global__ __launch_bounds__(256) void write_out(const float* __restrict__ act,
                                                 float* __restrict__ out) {
    int idx = blockIdx.x * blockDim.x + threadIdx.x;
    if (idx >= BD * 2 * DD) return;
    int b = idx >> 8, j = idx & (2 * DD - 1);
    out[idx] = act[(size_t)b * KD + j];
}

// ---------------------------------------------------------------------------
extern "C" void kernel_launch(void* const* d_in, const int* in_sizes, int n_in,
                              void* d_out, int out_size, void* d_ws, size_t ws_size,
                              hipStream_t stream) {
    const float* x     = (const float*)d_in[0];   // [N,128]
    const int*   batch = (const int*)d_in[1];     // [N] sorted
    const float* W_ih  = (const float*)d_in[2];   // [512,256]
    const float* W_hh  = (const float*)d_in[3];   // [512,128]
    const float* b_ih  = (const float*)d_in[4];   // [512]
    const float* b_hh  = (const float*)d_in[5];   // [512]
    float* out = (float*)d_out;                   // [2048,256]
    const int N = in_sizes[1];

    // workspace layout (floats)
    float* act   = (float*)d_ws;                  // 2048*384
    float* gates = act + (size_t)BD * KD;         // 2048*512
    float* cbuf  = gates + (size_t)BD * NDIM;     // 2048*128
    float* ebuf  = cbuf + (size_t)BD * DD;        // N (rounded up)
    float* WcT   = ebuf + 524288;                 // 384*512
    float* bsum  = WcT + (size_t)KD * NDIM;       // 512
    int*   seg   = (int*)(bsum + NDIM);           // 2049

    const int TB = 256;
    zero_state<<<(BD * KD + TB - 1) / TB, TB, 0, stream>>>(act, cbuf);
    prep_weights<<<(NDIM * KD + TB - 1) / TB, TB, 0, stream>>>(W_ih, W_hh, b_ih, b_hh, WcT, bsum);
    seg_bounds<<<(N + TB - 1) / TB, TB, 0, stream>>>(batch, seg, N);

    for (int t = 0; t < TT; ++t) {
        gemm_gates<<<(BD / 16) * (NDIM / 16) / 8, TB, 0, stream>>>(act, WcT, bsum, gates);
        lstm_pointwise<<<(BD * DD + TB - 1) / TB, TB, 0, stream>>>(gates, cbuf, act);
        attention_pool<<<BD, TB, 0, stream>>>(x, seg, act, ebuf);
    }
    write_out<<<(BD * 2 * DD + TB - 1) / TB, TB, 0, stream>>>(act, out);
}